// ChamferDistance_28621662060807
// MI455X (gfx1250) — compile-verified
//
#include <hip/hip_runtime.h>

typedef __attribute__((ext_vector_type(2))) float v2f;
typedef __attribute__((ext_vector_type(8))) float v8f;

#define BLOCK_THREADS 256
#define WAVES_PER_BLOCK 8
#define MAX_DST 4096

// For each src point i in batch b: out[b][i] = min_j || src[b][i] - dst[b][j] ||^2
//
// One wave handles a 16-row src tile via V_WMMA_F32_16X16X4_F32:
//   A row n  = (x1, y1, z1, 1)
//   B col m  = (-2*x2, -2*y2, -2*z2, ||p2||^2)   (precomputed once into LDS)
//   D[n][m]  = ||p2||^2 - 2*<p1,p2>
// dist[n][m] = D[n][m] + ||p1||^2 ; the row-constant ||p1||^2 is added once
// after the min-reduction over all m.
// Inner loop per tile = 1 ds_load_b64 (B fragment) + 1 WMMA + 4 v_min3.
__global__ __launch_bounds__(BLOCK_THREADS) void chamfer_dir_kernel(
    const float* __restrict__ src, const float* __restrict__ dst,
    float* __restrict__ out, int n_src, int n_dst)
{
    // Packed dst batch: per point (-2x, -2y, -2z, ||p||^2) -> 64 KB
    __shared__ float4 s_pk[MAX_DST];

    const int b   = blockIdx.y;
    const int tid = threadIdx.x;

    // ---- Stage + transform dst[b] into LDS once per block ----
    {
        const float* gb = dst + (size_t)b * n_dst * 3;
        for (int i = tid; i < n_dst; i += BLOCK_THREADS) {
            const float x = gb[3 * i + 0];
            const float y = gb[3 * i + 1];
            const float z = gb[3 * i + 2];
            s_pk[i] = make_float4(-2.0f * x, -2.0f * y, -2.0f * z,
                                  x * x + y * y + z * z);
        }
    }
    __syncthreads();

    const int wave = tid >> 5;         // wave32
    const int lane = tid & 31;
    const int li   = lane & 15;        // row/col index within the 16x16 tile
    const int half = lane >> 4;        // K-striping half (K0,K1 vs K2,K3)

    const int rowBase = (blockIdx.x * WAVES_PER_BLOCK + wave) * 16;
    const int row     = rowBase + li;

    // ---- Build A fragment: A[row] = (x1, y1, z1, 1) ----
    const float* sp = src + ((size_t)b * n_src + row) * 3;
    const float x1 = sp[0], y1 = sp[1], z1 = sp[2];
    const float sq1 = x1 * x1 + y1 * y1 + z1 * z1;

    v2f a;
    a[0] = half ? z1   : x1;   // VGPR0: K0 (lanes 0-15) / K2 (lanes 16-31)
    a[1] = half ? 1.0f : y1;   // VGPR1: K1 / K3

    float acc[8];
#pragma unroll
    for (int v = 0; v < 8; ++v) acc[v] = 3.0e38f;

    // B fragment per lane: float2 at element index t*32 + 2*li + half
    // (byte address t*256 + li*16 + half*8 -> base VGPR + imm offset per unroll)
    const float2* s2p = reinterpret_cast<const float2*>(s_pk);
    const int laneBase = 2 * li + half;

    // ---- Stream dst tiles; fixed 4x inner unroll, no remainder path ----
    const int ngroups = n_dst >> 6;    // groups of 4 tiles (n_dst % 64 == 0)
    for (int g = 0; g < ngroups; ++g) {
#pragma unroll
        for (int u = 0; u < 4; ++u) {
            const int t = 4 * g + u;
            const float2 bv = s2p[t * 32 + laneBase];
            v2f bb;
            bb[0] = bv.x;              // K0:-2x2 / K2:-2z2
            bb[1] = bv.y;              // K1:-2y2 / K3:sq2

            v8f c = {};
            v8f d = __builtin_amdgcn_wmma_f32_16x16x4_f32(
                /*neg_a=*/false, a, /*neg_b=*/false, bb,
                /*c_mod=*/(short)0, c, /*reuse_a=*/false, /*reuse_b=*/false);

#pragma unroll
            for (int v = 0; v < 8; ++v) acc[v] = fminf(acc[v], d[v]);
        }
    }

    // ---- Min across the 16 lanes of each half (rows v / v+8) ----
#pragma unroll
    for (int v = 0; v < 8; ++v) {
        float m = acc[v];
        m = fminf(m, __shfl_xor(m, 1, 16));
        m = fminf(m, __shfl_xor(m, 2, 16));
        m = fminf(m, __shfl_xor(m, 4, 16));
        m = fminf(m, __shfl_xor(m, 8, 16));
        acc[v] = m;
    }

    // ---- Write: lane whose own src row matches adds its sq1 and stores ----
    // half0 lanes 0..7 own rows v; half1 lanes 24..31 own rows v+8.
#pragma unroll
    for (int v = 0; v < 8; ++v) {
        const int r = v + 8 * half;
        if (li == r) out[(size_t)b * n_src + rowBase + r] = acc[v] + sq1;
    }
}

extern "C" void kernel_launch(void* const* d_in, const int* in_sizes, int n_in,
                              void* d_out, int out_size, void* d_ws, size_t ws_size,
                              hipStream_t stream) {
    const float* xyz1 = (const float*)d_in[0];
    const float* xyz2 = (const float*)d_in[1];

    const int B = 16;
    const int N = in_sizes[0] / (B * 3);   // 4096
    const int M = in_sizes[1] / (B * 3);   // 4096

    float* dist1 = (float*)d_out;                  // [B, N]
    float* dist2 = dist1 + (size_t)B * N;          // [B, M]

    dim3 block(BLOCK_THREADS);
    dim3 grid1(N / (WAVES_PER_BLOCK * 16), B);     // 32 x 16
    dim3 grid2(M / (WAVES_PER_BLOCK * 16), B);

    // Direction 1: for each xyz1 point, min sq-dist to xyz2.
    chamfer_dir_kernel<<<grid1, block, 0, stream>>>(xyz1, xyz2, dist1, N, M);
    // Direction 2: roles swapped (recompute instead of cross-block atomics).
    chamfer_dir_kernel<<<grid2, block, 0, stream>>>(xyz2, xyz1, dist2, M, N);
}